// RoiPooling_31731218383217
// MI455X (gfx1250) — compile-verified
//
#include <hip/hip_runtime.h>

#define POOL      14
#define FM_H      50
#define FM_W      50
#define FM_C      512
#define HALF_C    256          // channels per pass
#define HC4       64           // float4 chunks per 256-channel slice
#define NUM_ROIS  300
#define MAXW      25           // rois: w,h in [1,26)

// TF2 bilinear resize source coords for one axis (local, start not added):
// src = (i+0.5)*size/POOL - 0.5 ; lo = floor(src) ; frac = src-lo ;
// lo_i = clip(lo,0,size-1) ; hi_i = clip(lo+1,0,size-1)
__device__ __forceinline__ void axis_coords(int size, int i,
                                            int& lo, int& hi, float& frac) {
    float src = ((float)i + 0.5f) * (float)size / (float)POOL - 0.5f;
    float fl  = floorf(src);
    frac = src - fl;
    int l   = (int)fl;
    int sm1 = size - 1;
    lo = min(max(l, 0), sm1);
    hi = min(max(l + 1, 0), sm1);
}

// One block per (roi, output-row py). Stage feature rows y0,y1 (ROI x-span,
// 256 channels per pass) into LDS with gfx1250 async global->LDS copies,
// then bilinear-lerp 14 px x 256 ch with ds_load_b128 taps + b128 stores.
extern "C" __global__ __launch_bounds__(256)
void roi_pool_kernel(const float* __restrict__ fm,
                     const int*   __restrict__ rois,
                     float*       __restrict__ out) {
    extern __shared__ float smem[];            // [2][MAXW][HALF_C] floats
    const int blk = blockIdx.x;
    const int r   = blk / POOL;
    const int py  = blk - r * POOL;
    const int tid = threadIdx.x;

    const int rx = rois[r * 4 + 0];
    const int ry = rois[r * 4 + 1];
    const int rw = rois[r * 4 + 2];
    const int rh = rois[r * 4 + 3];
    const int w  = rw > MAXW ? MAXW : rw;      // safety clamp (harness: rw<=25)

    int y0, y1; float ty;
    axis_coords(rh, py, y0, y1, ty);
    y0 += ry; y1 += ry;

    const unsigned lds_base  = (unsigned)(uintptr_t)smem;  // low 32 bits = LDS offset
    const int      row_chunk = w * HC4;                    // float4 chunks per row slice
    const float4*  sm4       = reinterpret_cast<const float4*>(smem);

    const int quarter = tid >> 6;              // 0..3 -> pixel within group of 4
    const int c4      = tid & (HC4 - 1);       // float4 index within 256-ch slice
    float* outp = out + (((size_t)r * POOL + py) * POOL) * FM_C;

    for (int pass = 0; pass < 2; ++pass) {
        const int c_off = pass * HALF_C;
        if (pass) __syncthreads();             // LDS reads of pass 0 done

        // ---- async stage: rows y0,y1, columns [rx, rx+w), channels [c_off, c_off+256)
        for (int row = 0; row < 2; ++row) {
            const int yy = row ? y1 : y0;
            const float* rowp = fm + ((size_t)yy * FM_W + rx) * FM_C + c_off;
            const unsigned lrow = lds_base + (unsigned)(row * row_chunk) * 16u;
            for (int f = tid; f < row_chunk; f += 256) {
                const int xi = f >> 6;                 // column within ROI span
                const int cc = f & (HC4 - 1);          // float4 within slice
                const float* gp = rowp + (size_t)xi * FM_C + cc * 4;
                const unsigned laddr = lrow + (unsigned)f * 16u;
                asm volatile("global_load_async_to_lds_b128 %0, %1, off"
                             :: "v"(laddr), "v"((unsigned long long)(uintptr_t)gp)
                             : "memory");
            }
        }
        asm volatile("s_wait_asynccnt 0" ::: "memory");
        __syncthreads();

        // ---- compute: 4 px per iteration (64 lanes x float4 channels each)
        for (int pxb = 0; pxb < 16; pxb += 4) {
            const int px = pxb + quarter;
            if (px >= POOL) break;             // only quarters 2,3 at pxb==12
            int x0, x1; float tx;
            axis_coords(rw, px, x0, x1, tx);   // local coords (span starts at rx)

            float4 v00 = sm4[(x0)     * HC4 + c4];
            float4 v01 = sm4[(x1)     * HC4 + c4];
            float4 v10 = sm4[(w + x0) * HC4 + c4];
            float4 v11 = sm4[(w + x1) * HC4 + c4];

            float4 o;
            float top, bot;
            top = v00.x + (v01.x - v00.x) * tx; bot = v10.x + (v11.x - v10.x) * tx; o.x = top + (bot - top) * ty;
            top = v00.y + (v01.y - v00.y) * tx; bot = v10.y + (v11.y - v10.y) * tx; o.y = top + (bot - top) * ty;
            top = v00.z + (v01.z - v00.z) * tx; bot = v10.z + (v11.z - v10.z) * tx; o.z = top + (bot - top) * ty;
            top = v00.w + (v01.w - v00.w) * tx; bot = v10.w + (v11.w - v10.w) * tx; o.w = top + (bot - top) * ty;

            reinterpret_cast<float4*>(outp + (size_t)px * FM_C + c_off)[c4] = o;
        }
    }
}

extern "C" void kernel_launch(void* const* d_in, const int* in_sizes, int n_in,
                              void* d_out, int out_size, void* d_ws, size_t ws_size,
                              hipStream_t stream) {
    (void)in_sizes; (void)n_in; (void)out_size; (void)d_ws; (void)ws_size;
    const float* fm   = (const float*)d_in[0];   // [1,50,50,512] f32
    const int*   rois = (const int*)d_in[1];     // [1,300,4] i32 (x,y,w,h)
    float*       out  = (float*)d_out;           // [1,300,14,14,512] f32

    const dim3 grid(NUM_ROIS * POOL);
    const dim3 block(256);
    const size_t smem_bytes = 2u * MAXW * HALF_C * sizeof(float);  // 51,200 B
    hipLaunchKernelGGL(roi_pool_kernel, grid, block, smem_bytes, stream,
                       fm, rois, out);
}